// GPUFrequencyMask_38010460570422
// MI455X (gfx1250) — compile-verified
//
// CDNA5 (gfx1250) frequency-band masking via tensor-core FFT.
// y = irfft(keep * rfft(x)), x: [64, 8, 131072] fp32.
// Real signal packed as complex N=65536 = 256x256 four-step FFT; each
// 256-pt FFT = two complex 16x16 DFT matmuls on V_WMMA_F32_16X16X4_F32.
#include <hip/hip_runtime.h>
#include <math.h>

typedef float v2f __attribute__((ext_vector_type(2)));
typedef float v8f __attribute__((ext_vector_type(8)));

#define NC      65536          // packed complex length (= T/2)
#define NBINS   65537          // rfft bins of length-131072 signal
#define TWO_PI  6.28318530717958647692f

// ---- 16x16x16 fp32 matmul, chained over K in 4-wide chunks --------------
static __device__ __forceinline__ v8f mm16(const float a[8], const float b[8], v8f acc) {
#pragma unroll
  for (int kk = 0; kk < 4; ++kk) {
    v2f av; av.x = a[kk * 2 + 0]; av.y = a[kk * 2 + 1];
    v2f bv; bv.x = b[kk * 2 + 0]; bv.y = b[kk * 2 + 1];
    acc = __builtin_amdgcn_wmma_f32_16x16x4_f32(false, av, false, bv,
                                                (short)0, acc, false, false);
  }
  return acc;
}

// ---- per-wave 256-point (i)FFT ------------------------------------------
// Input: B-layout regs mr/mi of the 16x16 matrix M[a][b] = in[a*16+b].
// Output: D-layout outR/outI; element e of lane L is spectrum index
//         (L&15)*16 + e + 8*(L>>4)  (natural order after digit transpose).
// scratch: wave-private 256 x float2 region (contents destroyed).
template <bool INV>
static __device__ __forceinline__ void fft256_core(const float mr[8], const float mi[8],
                                                   float2* scratch, int lane,
                                                   v8f& outR, v8f& outI) {
  const float sgn  = INV ? 1.0f : -1.0f;
  const int   half = lane >> 4;
  const int   l15  = lane & 15;
  // DFT16 matrix constants in A-layout (16x4 chunks) and B-layout (4x16 chunks)
  float fAre[8], fAim[8], fBre[8], fBim[8];
#pragma unroll
  for (int kk = 0; kk < 4; ++kk) {
#pragma unroll
    for (int v = 0; v < 2; ++v) {
      int ka = kk * 4 + 2 * half + v;                 // A: M=l15, K=ka
      int ta = (l15 * ka) & 15;                       // exact angle reduction
      float sa, ca; __sincosf(sgn * (TWO_PI / 16.0f) * (float)ta, &sa, &ca);
      fAre[kk * 2 + v] = ca; fAim[kk * 2 + v] = sa;
      int rb = kk * 4 + v + 2 * half;                 // B: K=rb, N=l15
      int tb = (rb * l15) & 15;
      float sb, cb; __sincosf(sgn * (TWO_PI / 16.0f) * (float)tb, &sb, &cb);
      fBre[kk * 2 + v] = cb; fBim[kk * 2 + v] = sb;
    }
  }
  const v8f zero = {0.f, 0.f, 0.f, 0.f, 0.f, 0.f, 0.f, 0.f};
  // stage 1: S1 = F16 * M   (complex via 4 real WMMA chains)
  v8f rr = mm16(fAre, mr, zero);
  v8f ii = mm16(fAim, mi, zero);
  v8f ri = mm16(fAre, mi, zero);
  v8f ir = mm16(fAim, mr, zero);
  v8f s1r = rr - ii, s1i = ri + ir;
  // mid twiddle w256^(j1*b), bounce through LDS to transpose D-layout -> A-layout
#pragma unroll
  for (int e = 0; e < 8; ++e) {
    int j1 = e + half * 8;
    int t  = (j1 * l15) & 255;
    float s, c; __sincosf(sgn * (TWO_PI / 256.0f) * (float)t, &s, &c);
    float xr = s1r[e], xi = s1i[e];
    scratch[j1 * 16 + l15] = make_float2(xr * c - xi * s, xr * s + xi * c);
  }
  asm volatile("s_wait_dscnt 0" ::: "memory");   // wave-private LDS RAW fence
  float ar[8], ai[8];
#pragma unroll
  for (int kk = 0; kk < 4; ++kk) {
#pragma unroll
    for (int v = 0; v < 2; ++v) {
      int kc = kk * 4 + 2 * half + v;
      float2 t = scratch[l15 * 16 + kc];
      ar[kk * 2 + v] = t.x; ai[kk * 2 + v] = t.y;
    }
  }
  // stage 2: S3 = S2 * F16
  rr = mm16(ar, fBre, zero);
  ii = mm16(ai, fBim, zero);
  ri = mm16(ar, fBim, zero);
  ir = mm16(ai, fBre, zero);
  outR = rr - ii;
  outI = ri + ir;
}

// ---- rfft-bin mask helpers ----------------------------------------------
static __device__ __forceinline__ float keepf(int k, int s0, int e0, int s1, int e1) {
  bool m = (k >= s0 && k < e0) || (k >= s1 && k < e1);
  return m ? 0.0f : 1.0f;
}

// Given Zk = Z[k], Zp = Z[N-k] (packed-complex FFT values, N=65536), apply
// the rfft split, mask bins k and N-k, and recombine to Z'[k], Z'[N-k].
static __device__ __forceinline__ void mask_pair(float2 Zk, float2 Zp, int k,
                                                 float keepA, float keepB,
                                                 float2& zOut, float2& pOut) {
  float hsr = 0.5f * (Zk.x + Zp.x), hsi = 0.5f * (Zk.y - Zp.y);  // (Z + conj Zp)/2
  float hdr = 0.5f * (Zk.x - Zp.x), hdi = 0.5f * (Zk.y + Zp.y);  // (Z - conj Zp)/2
  float sw, cw; __sincosf(-(TWO_PI / 131072.0f) * (float)k, &sw, &cw); // W=e^{-i pi k/N}
  float Xar = hsr + sw * hdr + cw * hdi;            // X[k]   = hs + (sw - i cw) hd
  float Xai = hsi + sw * hdi - cw * hdr;
  float Xbr = hsr - (sw * hdr + cw * hdi);          // X[N-k] = conj(hs) - (sw + i cw) conj(hd)
  float Xbi = -hsi - (cw * hdr - sw * hdi);
  Xar *= keepA; Xai *= keepA;
  Xbr *= keepB; Xbi *= keepB;
  float Er = 0.5f * (Xar + Xbr), Ei = 0.5f * (Xai - Xbi);
  float Dr = 0.5f * (Xar - Xbr), Di = 0.5f * (Xai + Xbi);
  float Or = cw * Dr + sw * Di,  Oi = cw * Di - sw * Dr;          // O = conj(W) D
  zOut = make_float2(Er - Oi, Ei + Or);             // Z'[k]   = E + iO
  pOut = make_float2(Er + Oi, Or - Ei);             // Z'[N-k] = conj(E) + i conj(O)
}

// ---- Kernel A: pack + column FFT-256 (along n1) + big twiddle -----------
__global__ __launch_bounds__(256) void kA(const float* __restrict__ x,
                                          float2* __restrict__ ybuf) {
  __shared__ float2 tile[16 * 257];            // 16 columns, padded stride
  const int bc     = blockIdx.x >> 4;
  const int n2base = (blockIdx.x & 15) << 4;
  const int tid    = threadIdx.x;
  const float2* xin = (const float2*)x + (size_t)bc * NC;  // z[n]=x[2n]+ix[2n+1]
#pragma unroll
  for (int i = 0; i < 16; ++i) {               // coalesced 128B row chunks
    int n1 = i * 16 + (tid >> 4), n2l = tid & 15;
    tile[n2l * 257 + n1] = xin[n1 * 256 + n2base + n2l];
  }
  __syncthreads();
  const int wave = tid >> 5, lane = tid & 31;
  const int half = lane >> 4, l15 = lane & 15;
#pragma unroll
  for (int ci = 0; ci < 2; ++ci) {
    int col = wave * 2 + ci;
    float2* lbuf = &tile[col * 257];
    float mr[8], mi[8];
#pragma unroll
    for (int kk = 0; kk < 4; ++kk)
#pragma unroll
      for (int v = 0; v < 2; ++v) {
        float2 t = lbuf[(kk * 4 + v + 2 * half) * 16 + l15];
        mr[kk * 2 + v] = t.x; mi[kk * 2 + v] = t.y;
      }
    v8f oR, oI;
    fft256_core<false>(mr, mi, lbuf, lane, oR, oI);
    int n2g = n2base + col;
#pragma unroll
    for (int e = 0; e < 8; ++e) {              // twiddle w65536^(n2*k1), natural order
      int k1 = l15 * 16 + e + half * 8;
      int t  = (n2g * k1) & 65535;
      float s, c; __sincosf(-(TWO_PI / 65536.0f) * (float)t, &s, &c);
      float xr = oR[e], xi = oI[e];
      lbuf[k1] = make_float2(xr * c - xi * s, xr * s + xi * c);
    }
  }
  __syncthreads();
#pragma unroll
  for (int i = 0; i < 16; ++i) {               // coalesced store of Y[k1][n2]
    int k1 = i * 16 + (tid >> 4), n2l = tid & 15;
    ybuf[(size_t)bc * NC + k1 * 256 + n2base + n2l] = tile[n2l * 257 + k1];
  }
}

// ---- Kernel B: row FFT-256 + mask + inverse row FFT-256 (in place) ------
__global__ __launch_bounds__(64) void kB(float2* __restrict__ ybuf,
                                         const int* __restrict__ widths,
                                         const int* __restrict__ starts) {
  __shared__ float2 rowbuf[2][256];
  const int pid = blockIdx.x % 129;            // row-pair id 0..128
  const int bc  = blockIdx.x / 129;
  const int r   = pid;
  const int rp  = (256 - r) & 255;
  const bool selfpair = (rp == r);             // r==0 or r==128
  const int tid  = threadIdx.x;
  const int wave = tid >> 5, lane = tid & 31;
  const int half = lane >> 4, l15 = lane & 31 & 15;
  const int myrow = wave ? rp : r;
  const bool act  = (wave == 0) || !selfpair;  // uniform per-wave (EXEC stays full)
  float2* Y = ybuf + (size_t)bc * NC;

  if (act) {                                   // forward FFT of my row
    const float2* src = Y + myrow * 256;
    float mr[8], mi[8];
#pragma unroll
    for (int kk = 0; kk < 4; ++kk)
#pragma unroll
      for (int v = 0; v < 2; ++v) {
        float2 t = src[(kk * 4 + v + 2 * half) * 16 + l15];
        mr[kk * 2 + v] = t.x; mi[kk * 2 + v] = t.y;
      }
    v8f oR, oI;
    fft256_core<false>(mr, mi, rowbuf[wave], lane, oR, oI);
#pragma unroll
    for (int e = 0; e < 8; ++e)
      rowbuf[wave][l15 * 16 + e + half * 8] = make_float2(oR[e], oI[e]);
  }
  __syncthreads();

  // ---- mask + recombine: bin k = k2*256 + r pairs with N-k ----
  const int b  = bc >> 3;                      // C = 8
  const int s0 = starts[b],      e0 = min(starts[b] + widths[b], NBINS);
  const int s1 = starts[64 + b], e1 = min(starts[64 + b] + widths[64 + b], NBINS);
  if (!selfpair) {
    for (int k2 = tid; k2 < 256; k2 += 64) {
      int k2p = 255 - k2;
      int k   = k2 * 256 + r;
      float2 zo, po;
      mask_pair(rowbuf[0][k2], rowbuf[1][k2p], k,
                keepf(k, s0, e0, s1, e1), keepf(65536 - k, s0, e0, s1, e1), zo, po);
      rowbuf[0][k2]  = zo;
      rowbuf[1][k2p] = po;
    }
  } else if (r == 0) {
    for (int k2 = tid; k2 <= 128; k2 += 64) {
      if (k2 == 0) {                           // DC and true Nyquist (bin 65536)
        float2 Z0 = rowbuf[0][0];
        float a0 = keepf(0, s0, e0, s1, e1)     * (Z0.x + Z0.y);
        float aN = keepf(65536, s0, e0, s1, e1) * (Z0.x - Z0.y);
        rowbuf[0][0] = make_float2(0.5f * (a0 + aN), 0.5f * (a0 - aN));
      } else {
        int k2p = 256 - k2;                    // pair inside row 0 (k2==128 self-consistent)
        int k   = k2 * 256;
        float2 zo, po;
        mask_pair(rowbuf[0][k2], rowbuf[0][k2p], k,
                  keepf(k, s0, e0, s1, e1), keepf(65536 - k, s0, e0, s1, e1), zo, po);
        rowbuf[0][k2]  = zo;
        rowbuf[0][k2p] = po;
      }
    }
  } else {                                     // r == 128: pairs (k2, 255-k2) in row
    for (int k2 = tid; k2 < 128; k2 += 64) {
      int k2p = 255 - k2;
      int k   = k2 * 256 + 128;
      float2 zo, po;
      mask_pair(rowbuf[0][k2], rowbuf[0][k2p], k,
                keepf(k, s0, e0, s1, e1), keepf(65536 - k, s0, e0, s1, e1), zo, po);
      rowbuf[0][k2]  = zo;
      rowbuf[0][k2p] = po;
    }
  }
  __syncthreads();

  if (act) {                                   // inverse stage 1 + inverse twiddle
    float mr[8], mi[8];
#pragma unroll
    for (int kk = 0; kk < 4; ++kk)
#pragma unroll
      for (int v = 0; v < 2; ++v) {
        float2 t = rowbuf[wave][(kk * 4 + v + 2 * half) * 16 + l15];
        mr[kk * 2 + v] = t.x; mi[kk * 2 + v] = t.y;
      }
    v8f oR, oI;
    fft256_core<true>(mr, mi, rowbuf[wave], lane, oR, oI);
#pragma unroll
    for (int e = 0; e < 8; ++e) {
      int n2 = l15 * 16 + e + half * 8;
      int t  = (n2 * myrow) & 65535;
      float s, c; __sincosf((TWO_PI / 65536.0f) * (float)t, &s, &c);
      float xr = oR[e], xi = oI[e];
      rowbuf[wave][n2] = make_float2(xr * c - xi * s, xr * s + xi * c);
    }
    asm volatile("s_wait_dscnt 0" ::: "memory");
#pragma unroll
    for (int j = 0; j < 8; ++j) {              // coalesced row writeback (in place)
      int idx = j * 32 + lane;
      Y[myrow * 256 + idx] = rowbuf[wave][idx];
    }
  }
}

// ---- Kernel C: column iFFT-256 (along k1) + 1/N scale + unpack ----------
__global__ __launch_bounds__(256) void kC(const float2* __restrict__ wbuf,
                                          float* __restrict__ out) {
  __shared__ float2 tile[16 * 257];
  const int bc     = blockIdx.x >> 4;
  const int n2base = (blockIdx.x & 15) << 4;
  const int tid    = threadIdx.x;
  const float2* src = wbuf + (size_t)bc * NC;
#pragma unroll
  for (int i = 0; i < 16; ++i) {
    int k1 = i * 16 + (tid >> 4), n2l = tid & 15;
    tile[n2l * 257 + k1] = src[k1 * 256 + n2base + n2l];
  }
  __syncthreads();
  const int wave = tid >> 5, lane = tid & 31;
  const int half = lane >> 4, l15 = lane & 15;
#pragma unroll
  for (int ci = 0; ci < 2; ++ci) {
    int col = wave * 2 + ci;
    float2* lbuf = &tile[col * 257];
    float mr[8], mi[8];
#pragma unroll
    for (int kk = 0; kk < 4; ++kk)
#pragma unroll
      for (int v = 0; v < 2; ++v) {
        float2 t = lbuf[(kk * 4 + v + 2 * half) * 16 + l15];
        mr[kk * 2 + v] = t.x; mi[kk * 2 + v] = t.y;
      }
    v8f oR, oI;
    fft256_core<true>(mr, mi, lbuf, lane, oR, oI);
#pragma unroll
    for (int e = 0; e < 8; ++e) {
      int n1 = l15 * 16 + e + half * 8;
      lbuf[n1] = make_float2(oR[e], oI[e]);
    }
  }
  __syncthreads();
  float2* o = (float2*)out + (size_t)bc * NC;  // x'[2n]=Re z', x'[2n+1]=Im z'
  const float sc = 1.0f / 65536.0f;
#pragma unroll
  for (int i = 0; i < 16; ++i) {
    int n1 = i * 16 + (tid >> 4), n2l = tid & 15;
    float2 v = tile[n2l * 257 + n1];
    o[n1 * 256 + n2base + n2l] = make_float2(v.x * sc, v.y * sc);
  }
}

extern "C" void kernel_launch(void* const* d_in, const int* in_sizes, int n_in,
                              void* d_out, int out_size, void* d_ws, size_t ws_size,
                              hipStream_t stream) {
  (void)in_sizes; (void)n_in; (void)out_size; (void)ws_size;
  const float* x      = (const float*)d_in[0];
  const int*   widths = (const int*)d_in[1];   // [2][64]
  const int*   starts = (const int*)d_in[2];   // [2][64]
  float2* ybuf = (float2*)d_ws;                // 512 * 65536 complex = 256 MB scratch
  const int BC = 64 * 8;
  kA<<<dim3(BC * 16), 256, 0, stream>>>(x, ybuf);
  kB<<<dim3(BC * 129), 64, 0, stream>>>(ybuf, widths, starts);
  kC<<<dim3(BC * 16), 256, 0, stream>>>(ybuf, (float*)d_out);
}